// ProcessGNN_33311766347862
// MI455X (gfx1250) — compile-verified
//
#include <hip/hip_runtime.h>

typedef __attribute__((ext_vector_type(2))) float v2f;
typedef __attribute__((ext_vector_type(8))) float v8f;
typedef __attribute__((ext_vector_type(4))) unsigned int u32x4;
typedef __attribute__((ext_vector_type(8))) int i32x8;
typedef __attribute__((ext_vector_type(4))) int i32x4;

#define HID 64

// ---------------------------------------------------------------------------
// Degree / normalization: deg[i] = 1 (self loop) + #targets == i; dinv = rsqrt
// ---------------------------------------------------------------------------
__global__ void deg_init_kernel(float* __restrict__ dinv, int n) {
  int i = blockIdx.x * 256 + threadIdx.x;
  if (i < n) dinv[i] = 1.0f;
}

__global__ void deg_count_kernel(const int* __restrict__ col,
                                 float* __restrict__ dinv, int e) {
  int i = blockIdx.x * 256 + threadIdx.x;
  if (i < e) unsafeAtomicAdd(&dinv[col[i]], 1.0f);
}

__global__ void deg_rsqrt_kernel(float* __restrict__ dinv, int n) {
  int i = blockIdx.x * 256 + threadIdx.x;
  if (i < n) dinv[i] = rsqrtf(dinv[i]);
}

// ---------------------------------------------------------------------------
// out[n,64] = A[n,K] @ W[K,64] (+bias)  via V_WMMA_F32_16X16X4_F32.
// 256 threads = 8 waves; each wave -> one 16-row tile; block covers 128 rows.
// W (K*64 floats, contiguous) staged into LDS by the Tensor Data Mover:
// one TENSOR_LOAD_TO_LDS issued by wave 0, completion via s_wait_tensorcnt.
// ---------------------------------------------------------------------------
__global__ __launch_bounds__(256) void gemm_wmma_kernel(
    const float* __restrict__ A, const float* __restrict__ W,
    const float* __restrict__ bias, float* __restrict__ out, int n, int K) {
  __shared__ float Wlds[64 * 64];
  int tid = threadIdx.x;

  if (tid < 32) {  // wave 0 drives the TDM copy of W -> LDS
    unsigned elems = (unsigned)(K * 64);            // <= 4096, fits tile_dim0
    unsigned lds_addr = (unsigned)(size_t)(void*)&Wlds[0];  // addr[31:0] = LDS offset
    unsigned long long ga = (unsigned long long)(size_t)W;

    u32x4 g0;
    g0[0] = 1u;                                     // count=1, user descriptor
    g0[1] = lds_addr;                               // lds_addr [63:32]
    g0[2] = (unsigned)(ga & 0xffffffffu);           // global_addr lo
    g0[3] = ((unsigned)(ga >> 32) & 0x01ffffffu)    // global_addr hi (57-bit)
            | (2u << 30);                           // type = 2 ("image")

    i32x8 g1;
    g1[0] = (int)(2u << 16);                        // data_size=4B; mask=0; no pad
    g1[1] = (int)((elems & 0xffffu) << 16);         // tensor_dim0[15:0]
    g1[2] = (int)(((elems >> 16) & 0xffffu)         // tensor_dim0[31:16]
                  | (1u << 16));                    // tensor_dim1 = 1
    g1[3] = (int)((elems & 0xffffu) << 16);         // tensor_dim1 hi=0; tile_dim0=elems
    g1[4] = 0;                                      // tile_dim1/2 unused
    g1[5] = (int)elems;                             // tensor_dim0_stride lo
    g1[6] = 0;                                      // stride0 hi / stride1 lo
    g1[7] = 0;

    i32x4 g2;
    g2[0] = 1;                                      // tensor_dim2 = 1
    g2[1] = 1;                                      // tensor_dim3 = 1
    g2[2] = (int)elems;                             // tensor_dim2_stride lo
    g2[3] = 0;                                      // stride hi; tile_dim3 = 0

    i32x4 g3;
    g3[0] = (int)elems;                             // tensor_dim3_stride lo
    g3[1] = (int)(1u << 16);                        // tensor_dim4 = 1
    g3[2] = 0;                                      // tile_dim4 = 0
    g3[3] = 0;

    i32x8 g4 = {0, 0, 0, 0, 0, 0, 0, 0};            // extra operand (clang-23 form)

    __builtin_amdgcn_tensor_load_to_lds(g0, g1, g2, g3, g4, 0);
    __builtin_amdgcn_s_wait_tensorcnt(0);
  }
  __syncthreads();

  int wave = tid >> 5;
  int lane = tid & 31;
  int half = lane >> 4;   // 0 or 1
  int m    = lane & 15;   // row within tile / col within N-tile
  long rowbase = (long)blockIdx.x * 128 + wave * 16;

  // Clamp A row: garbage rows only feed D rows >= n, which are never stored.
  long arow = rowbase + m;
  if (arow >= n) arow = n - 1;
  const float* aptr = A + arow * K + 2 * half;

  v8f acc[4] = {v8f{}, v8f{}, v8f{}, v8f{}};
  int nsteps = K >> 2;
  for (int k = 0; k < nsteps; ++k) {
    int kb = k * 4;
    // A 16x4 frag: a.x = A[m][kb+2*half], a.y = A[m][kb+2*half+1]
    v2f a = *(const v2f*)(aptr + kb);
    int krow = (kb + 2 * half) * 64;  // B row offset in LDS
#pragma unroll
    for (int nt = 0; nt < 4; ++nt) {
      v2f b;
      b.x = Wlds[krow + nt * 16 + m];       // B[kb+2*half][nt*16+m]
      b.y = Wlds[krow + 64 + nt * 16 + m];  // B[kb+2*half+1][nt*16+m]
      acc[nt] = __builtin_amdgcn_wmma_f32_16x16x4_f32(
          false, a, false, b, (short)0, acc[nt], false, false);
    }
  }

#pragma unroll
  for (int nt = 0; nt < 4; ++nt) {
    float badd = bias ? bias[nt * 16 + m] : 0.0f;
#pragma unroll
    for (int r = 0; r < 8; ++r) {
      long row = rowbase + r + 8 * half;
      if (row < n) out[row * HID + nt * 16 + m] = acc[nt][r] + badd;
    }
  }
}

// ---------------------------------------------------------------------------
// hacc[i,:] = dinv[i]^2 * hw[i,:]   (self-loop message; pure write -> also
// (re)initializes the accumulator every call, workspace is poisoned)
// ---------------------------------------------------------------------------
__global__ void selfloop_kernel(const float* __restrict__ hw,
                                const float* __restrict__ dinv,
                                float* __restrict__ hacc, int n) {
  long idx = (long)blockIdx.x * 256 + threadIdx.x;  // over n*32 float2 slots
  if (idx >= (long)n * 32) return;
  int node = (int)(idx >> 5);
  int f2 = ((int)idx & 31) * 2;
  float d = dinv[node];
  float s = d * d;
  v2f v = *(const v2f*)(hw + (long)node * HID + f2);
  v2f o; o.x = v.x * s; o.y = v.y * s;
  *(v2f*)(hacc + (long)node * HID + f2) = o;
}

// ---------------------------------------------------------------------------
// Edge scatter: hacc[col[e],:] += dinv[row]*dinv[col] * hw[row[e],:]
// one wave per edge, one float2 per lane -> hw reads fully coalesced, and the
// 25.6MB node matrix stays L2-resident (192MB L2).
// ---------------------------------------------------------------------------
__global__ void scatter_kernel(const int* __restrict__ row,
                               const int* __restrict__ col,
                               const float* __restrict__ dinv,
                               const float* __restrict__ hw,
                               float* __restrict__ hacc, long e) {
  long idx = (long)blockIdx.x * 256 + threadIdx.x;  // over e*32 float2 slots
  if (idx >= e * 32) return;
  int eid = (int)(idx >> 5);
  int f2 = ((int)idx & 31) * 2;
  int r = row[eid];
  int c = col[eid];
  float s = dinv[r] * dinv[c];
  v2f v = *(const v2f*)(hw + (long)r * HID + f2);
  float* dst = hacc + (long)c * HID + f2;
  unsafeAtomicAdd(dst, v.x * s);
  unsafeAtomicAdd(dst + 1, v.y * s);
}

// ---------------------------------------------------------------------------
// h = relu(hacc + bias)
// ---------------------------------------------------------------------------
__global__ void bias_relu_kernel(const float* __restrict__ hacc,
                                 const float* __restrict__ bias,
                                 float* __restrict__ h, long total) {
  long idx = (long)blockIdx.x * 256 + threadIdx.x;
  if (idx >= total) return;
  float v = hacc[idx] + bias[(int)idx & 63];
  h[idx] = v > 0.0f ? v : 0.0f;
}

// ---------------------------------------------------------------------------
// Mean pool (sum stage): gsum[f] += sum over 512-row chunk of h[:,f]
// ---------------------------------------------------------------------------
__global__ void zero64_kernel(float* __restrict__ gsum) {
  if (threadIdx.x < 64) gsum[threadIdx.x] = 0.0f;
}

__global__ __launch_bounds__(256) void pool_kernel(const float* __restrict__ h,
                                                   float* __restrict__ gsum,
                                                   int n) {
  __shared__ float red[256];
  int t = threadIdx.x;
  int f = t & 63;
  int rl = t >> 6;  // 0..3
  long base = (long)blockIdx.x * 512;
  long lim = base + 512 < n ? base + 512 : n;
  float s = 0.0f;
  for (long r = base + rl; r < lim; r += 4) s += h[r * HID + f];
  red[t] = s;
  __syncthreads();
  if (t < 64) {
    float tot = red[t] + red[t + 64] + red[t + 128] + red[t + 192];
    unsafeAtomicAdd(&gsum[t], tot);
  }
}

// ---------------------------------------------------------------------------
// Head: out = relu(mean(h) @ W1 + b1) @ W2 + b2   (one wave32)
// ---------------------------------------------------------------------------
__global__ void head_kernel(const float* __restrict__ gsum,
                            const float* __restrict__ W1,
                            const float* __restrict__ b1,
                            const float* __restrict__ W2,
                            const float* __restrict__ b2,
                            float* __restrict__ out, int n) {
  __shared__ float red[32];
  int j = threadIdx.x;  // 0..31
  float invn = 1.0f / (float)n;
  float acc = b1[j];
#pragma unroll
  for (int f = 0; f < 64; ++f) acc += (gsum[f] * invn) * W1[f * 32 + j];
  float z = acc > 0.0f ? acc : 0.0f;
  red[j] = z * W2[j];
  __syncthreads();
  if (j == 0) {
    float s = 0.0f;
    for (int k = 0; k < 32; ++k) s += red[k];
    out[0] = s + b2[0];
  }
}

// ---------------------------------------------------------------------------
extern "C" void kernel_launch(void* const* d_in, const int* in_sizes, int n_in,
                              void* d_out, int out_size, void* d_ws,
                              size_t ws_size, hipStream_t stream) {
  const float* x     = (const float*)d_in[0];
  const int*   eidx  = (const int*)d_in[2];
  const float* W_enc = (const float*)d_in[3];
  const float* b_enc = (const float*)d_in[4];
  const float* Wg    = (const float*)d_in[7];
  const float* bg    = (const float*)d_in[8];
  const float* W1    = (const float*)d_in[9];
  const float* b1    = (const float*)d_in[10];
  const float* W2    = (const float*)d_in[11];
  const float* b2    = (const float*)d_in[12];

  int n = in_sizes[0] / 32;        // F_N = 32
  long e = in_sizes[2] / 2;        // edge_index is [2, E]
  const int* row = eidx;           // sources
  const int* col = eidx + e;       // targets

  float* ws = (float*)d_ws;
  long nh = (long)n * HID;
  long npad = ((long)n + 255) / 256 * 256;
  float* dinv = ws;
  float* h    = ws + npad;
  float* hw   = h + nh;
  float* hacc = hw + nh;
  float* gsum = hacc + nh;

  int nb256 = (n + 255) / 256;
  int eb256 = (int)((e + 255) / 256);
  int ggemm = (n + 127) / 128;
  int gself = (int)(((long)n * 32 + 255) / 256);
  int gscat = (int)((e * 32 + 255) / 256);
  int gelem = (int)((nh + 255) / 256);

  // symmetric normalization
  deg_init_kernel<<<nb256, 256, 0, stream>>>(dinv, n);
  deg_count_kernel<<<eb256, 256, 0, stream>>>(col, dinv, (int)e);
  deg_rsqrt_kernel<<<nb256, 256, 0, stream>>>(dinv, n);

  // node encoder: h = x @ W_enc + b_enc   (edge encoder is dead code)
  gemm_wmma_kernel<<<ggemm, 256, 0, stream>>>(x, W_enc, b_enc, h, n, 32);

  // 3 GCN layers
  for (int l = 0; l < 3; ++l) {
    gemm_wmma_kernel<<<ggemm, 256, 0, stream>>>(h, Wg + l * 64 * 64, nullptr,
                                                hw, n, 64);
    selfloop_kernel<<<gself, 256, 0, stream>>>(hw, dinv, hacc, n);
    scatter_kernel<<<gscat, 256, 0, stream>>>(row, col, dinv, hw, hacc, e);
    bias_relu_kernel<<<gelem, 256, 0, stream>>>(hacc, bg + l * HID, h, nh);
  }

  // mean pool + MLP head
  zero64_kernel<<<1, 64, 0, stream>>>(gsum);
  pool_kernel<<<(n + 511) / 512, 256, 0, stream>>>(h, gsum, n);
  head_kernel<<<1, 32, 0, stream>>>(gsum, W1, b1, W2, b2, (float*)d_out, n);
}